// ReprojectionLayer_19731079758021
// MI455X (gfx1250) — compile-verified
//
#include <hip/hip_runtime.h>

typedef float v2f __attribute__((ext_vector_type(2)));
typedef float v8f __attribute__((ext_vector_type(8)));

#define GDIM 64
#define G3   (GDIM * GDIM * GDIM)   // 262144 voxels
#define NC   12
#define NJ   23
#define HMH  512
#define HMW  640
#define HW   (HMH * HMW)            // 327680

// One wave32 handles 32 voxels = two 16-voxel WMMA tiles.
// Per tile: 3x V_WMMA_F32_16X16X4_F32 computes proj[k][cam][voxel]:
//   A (16x4): row c = camera c, A[c][f] = cameraMatrices[c][f][k]  (rows 12..15 = 0)
//   B (4x16): col n = (px,py,pz,1) of voxel n
//   D (16x16): lane l (n=l&15) holds cams r+8*(l>>4) in VGPR r
// => lane n has cams 0..7, lane n+16 has cams 8..11 of the same voxel.
__global__ __launch_bounds__(256)
void ReprojectionLayer_kernel(const float* __restrict__ hm,      // [1,12,23,512,640]
                              const float* __restrict__ center,  // [1,3]
                              const float* __restrict__ cam,     // [12,4,3]
                              float* __restrict__ out)           // [1,23,64,64,64]
{
    const int lane = threadIdx.x & 31;
    const int wave = threadIdx.x >> 5;
    const int n    = lane & 15;       // voxel-in-tile (B/D column)
    const int half = lane >> 4;       // 0: K={0,1}, cams 0..7 ; 1: K={2,3}, cams 8..11

    const int base = (blockIdx.x * 8 + wave) * 32;   // first of 32 voxels for this wave

    const float cx = center[0], cy = center[1], cz = center[2];

    // ---- A operands (voxel-independent): row = camera (lane&15), K-half = 2*half
    const int kb = 2 * half;
    v2f a0 = {0.f, 0.f}, a1 = {0.f, 0.f}, a2 = {0.f, 0.f};
    if (n < NC) {
        const float* m = cam + n * 12;               // cameraMatrices[c] : [4][3]
        a0.x = m[kb * 3 + 0]; a0.y = m[(kb + 1) * 3 + 0];
        a1.x = m[kb * 3 + 1]; a1.y = m[(kb + 1) * 3 + 1];
        a2.x = m[kb * 3 + 2]; a2.y = m[(kb + 1) * 3 + 2];
    }

    // off[t][r] = (cam*23)*HW + pixel index, reused for all 23 joints.
    // Entries for half==1, r>=4 ("cams 12..15") hold garbage but are NEVER
    // dereferenced (j-loop guards them), so no select/branch is needed:
    // the math is lane-safe (zero A rows -> NaN -> maxmin clamps to 0).
    int off[2][8];

    #pragma unroll
    for (int t = 0; t < 2; ++t) {
        const int v  = base + t * 16 + n;            // this column's voxel id
        const int xi = v >> 12, yi = (v >> 6) & 63, zi = v & 63;
        const float px = (float)(xi - 32) * 2.0f + cx;
        const float py = (float)(yi - 32) * 2.0f + cy;
        const float pz = (float)(zi - 32) * 2.0f + cz;

        // B operand: VGPR0 = B[2*half][n], VGPR1 = B[2*half+1][n]
        v2f bm;
        bm.x = half ? pz : px;
        bm.y = half ? 1.0f : py;

        v8f zero = {};
        v8f d0 = __builtin_amdgcn_wmma_f32_16x16x4_f32(false, a0, false, bm, (short)0, zero, false, false);
        v8f d1 = __builtin_amdgcn_wmma_f32_16x16x4_f32(false, a1, false, bm, (short)0, zero, false, false);
        v8f d2 = __builtin_amdgcn_wmma_f32_16x16x4_f32(false, a2, false, bm, (short)0, zero, false, false);

        #pragma unroll
        for (int r = 0; r < 8; ++r) {
            const float p0 = d0[r], p1 = d1[r], p2 = d2[r];
            // fast reciprocal + one Newton step, folded with the *0.5
            float rinv = __builtin_amdgcn_rcpf(p2);
            rinv = __builtin_fmaf(__builtin_fmaf(-p2, rinv, 1.0f), rinv, rinv);
            const float rh = rinv * 0.5f;
            const float uh = fminf(fmaxf(p0 * rh, 0.0f), 639.5f);   // u/2 in [0, 639.5]
            const float vh = fminf(fmaxf(p1 * rh, 0.0f), 511.5f);   // v/2 in [0, 511.5]
            const int idx  = (int)vh * HMW + (int)uh;
            const int c    = r + 8 * half;
            off[t][r] = c * (NJ * HW) + idx;
        }
    }

    const float scale = 1.0f / 12.0f;
    for (int j = 0; j < NJ; ++j) {
        const int jb = j * HW;
        float accA = 0.0f, accB = 0.0f;
        #pragma unroll
        for (int r = 0; r < 4; ++r) {                // rows valid for both halves
            accA += hm[(size_t)(off[0][r] + jb)];
            accB += hm[(size_t)(off[1][r] + jb)];
        }
        if (half == 0) {                             // rows 4..7: cams 4..7 only
            #pragma unroll
            for (int r = 4; r < 8; ++r) {
                accA += hm[(size_t)(off[0][r] + jb)];
                accB += hm[(size_t)(off[1][r] + jb)];
            }
        }
        // combine cams 0..7 (lane n) with cams 8..11 (lane n+16)
        accA += __shfl_xor(accA, 16, 32);
        accB += __shfl_xor(accB, 16, 32);
        // lanes 0..15 own tile 0 voxels, lanes 16..31 own tile 1 -> 32 contiguous floats
        const float res = ((half == 0) ? accA : accB) * scale;
        // write-once output: non-temporal store keeps L2 for heatmap gathers
        __builtin_nontemporal_store(res, &out[(size_t)j * G3 + base + lane]);
    }
}

extern "C" void kernel_launch(void* const* d_in, const int* in_sizes, int n_in,
                              void* d_out, int out_size, void* d_ws, size_t ws_size,
                              hipStream_t stream) {
    (void)in_sizes; (void)n_in; (void)out_size; (void)d_ws; (void)ws_size;
    const float* hm     = (const float*)d_in[0];   // heatmaps  [1,12,23,512,640]
    const float* center = (const float*)d_in[1];   // center    [1,3]
    const float* cam    = (const float*)d_in[2];   // cameras   [12,4,3]
    float* out          = (float*)d_out;           // [1,23,64,64,64]

    // 262144 voxels / 32 per wave / 8 waves per block = 1024 blocks
    ReprojectionLayer_kernel<<<G3 / 256, 256, 0, stream>>>(hm, center, cam, out);
}